// MLSelectiveKD_54966991454809
// MI455X (gfx1250) — compile-verified
//
#include <hip/hip_runtime.h>

#define B_  256
#define C_  10000
#define K_  8
#define TPB 256

typedef float v2f __attribute__((ext_vector_type(2)));
typedef float v8f __attribute__((ext_vector_type(8)));

// Full 32-lane sum via the matrix pipe:
// A (16x4 f32): lane L<16 -> A[L][0]=v0, A[L][1]=v1 ; lane L>=16 -> A[L-16][2]=v0, A[L-16][3]=v1.
// With v0 = x, v1 = 0 and B = all-ones: D[m][n] = x[m] + x[m+16] for every n.
// D layout: lanes 0-15 hold rows 0-7 in VGPRs 0-7, lanes 16-31 hold rows 8-15.
// Sum of a lane's 8 D values + shfl_xor(16) == sum over all 32 lanes, in every lane.
__device__ __forceinline__ float wmma_wave_sum(float x) {
  v2f a; a.x = x;    a.y = 0.0f;
  v2f b; b.x = 1.0f; b.y = 1.0f;
  v8f c = {0.0f, 0.0f, 0.0f, 0.0f, 0.0f, 0.0f, 0.0f, 0.0f};
  c = __builtin_amdgcn_wmma_f32_16x16x4_f32(false, a, false, b, (short)0, c,
                                            false, false);
  float s = ((c[0] + c[1]) + (c[2] + c[3])) + ((c[4] + c[5]) + (c[6] + c[7]));
  s += __shfl_xor(s, 16, 32);
  return s;
}

// One block per sample row. Streams s/t once (float4 / B128), writes the s copy
// in the same pass, and accumulates:
//   zs = sum exp(s/T), zt = sum exp(t/T), w = sum exp(t/T)*(t/T - s/T),
//   kd = sum [ sig(t/T)*(t/T - s/T) + log((1+e^{s/T})/(1+e^{t/T})) ]
// (the Bernoulli-KL log1p pair collapses to one log; es,et are O(1) so no
// precision loss vs log1p). Thread 0 then folds in the K=8 positive-label
// corrections in closed form: masked logits exp to exactly 0, so each variant
// k's partition function is the full row sum minus the other 7 positives.
__global__ __launch_bounds__(TPB) void kd_row_kernel(
    const float* __restrict__ s, const float* __restrict__ t,
    const int* __restrict__ pos, float* __restrict__ out_s,
    float* __restrict__ ws) {
  const int b = blockIdx.x;
  const int tid = threadIdx.x;
  const float invT = 1.0f / 3.0f;

  const float4* s4 = (const float4*)(s + (size_t)b * C_);
  const float4* t4 = (const float4*)(t + (size_t)b * C_);
  float4* o4 = (float4*)(out_s + (size_t)b * C_);

  float zs = 0.0f, zt = 0.0f, w = 0.0f, kd = 0.0f;

  for (int i = tid; i < C_ / 4; i += TPB) {
    float4 sv = s4[i];
    float4 tv = t4[i];
    o4[i] = sv;  // fused copy of s_out -> output slot 0
#pragma unroll
    for (int c = 0; c < 4; ++c) {
      float a  = ((const float*)&sv)[c] * invT;
      float bt = ((const float*)&tv)[c] * invT;
      float es = __expf(a);
      float et = __expf(bt);
      float d  = bt - a;
      zs += es;
      zt += et;
      w  = __fmaf_rn(et, d, w);
      float r  = 1.0f / (1.0f + et);         // shared by sigmoid and log-ratio
      float tp = et * r;                     // sigmoid(t/T)
      kd += __fmaf_rn(tp, d, __logf((1.0f + es) * r));
    }
  }

  // EXEC is all-ones here (loop has reconverged) -> WMMA legal.
  float rzs = wmma_wave_sum(zs);
  float rzt = wmma_wave_sum(zt);
  float rw  = wmma_wave_sum(w);
  float rkd = wmma_wave_sum(kd);

  __shared__ float red[8][4];
  const int wave = tid >> 5, lane = tid & 31;
  if (lane == 0) {
    red[wave][0] = rzs; red[wave][1] = rzt;
    red[wave][2] = rw;  red[wave][3] = rkd;
  }
  __syncthreads();

  if (tid == 0) {
    float Zs = 0.0f, Zt = 0.0f, W = 0.0f, KD = 0.0f;
#pragma unroll
    for (int v = 0; v < 8; ++v) {
      Zs += red[v][0]; Zt += red[v][1]; W += red[v][2]; KD += red[v][3];
    }
    float es[K_], et[K_], wj[K_];
    float Es = 0.0f, Et = 0.0f, Wp = 0.0f;
#pragma unroll
    for (int j = 0; j < K_; ++j) {
      int p    = pos[b * K_ + j];
      float a  = s[(size_t)b * C_ + p] * invT;
      float bt = t[(size_t)b * C_ + p] * invT;
      es[j] = __expf(a);
      et[j] = __expf(bt);
      wj[j] = et[j] * (bt - a);
      Es += es[j]; Et += et[j]; Wp += wj[j];
    }
    float kl = 0.0f;
#pragma unroll
    for (int k = 0; k < K_; ++k) {
      float Zsk = Zs - Es + es[k];
      float Ztk = Zt - Et + et[k];
      kl += (W - Wp + wj[k]) / Ztk - (__logf(Ztk) - __logf(Zsk));
    }
    ws[b]      = kl;  // per-row selective-KD sum over its K labels
    ws[B_ + b] = KD;  // per-row dense sigmoid-KD sum
  }
}

// Single block: deterministically reduce the 256 per-row partials and emit the
// loss vector (loss[0] = loss0, rest zero). Also uses the WMMA reduction.
__global__ __launch_bounds__(TPB) void kd_final_kernel(
    const float* __restrict__ ws, float* __restrict__ loss) {
  const int tid = threadIdx.x;
  float sel = ws[tid];
  float kdp = ws[B_ + tid];

  float rsel = wmma_wave_sum(sel);
  float rkd  = wmma_wave_sum(kdp);

  __shared__ float red[8][2];
  const int wave = tid >> 5, lane = tid & 31;
  if (lane == 0) { red[wave][0] = rsel; red[wave][1] = rkd; }
  __syncthreads();

  float out = 0.0f;
  if (tid == 0) {
    float SEL = 0.0f, KD = 0.0f;
#pragma unroll
    for (int v = 0; v < 8; ++v) { SEL += red[v][0]; KD += red[v][1]; }
    const float T2 = 9.0f;          // TEMP^2
    const float ALPHA = 0.5f;
    out = SEL * (T2 / (float)K_) * ALPHA + KD * (T2 / (float)C_) * (1.0f - ALPHA);
  }
  loss[tid] = out;  // 256 threads cover all B loss entries (poison-safe)
}

extern "C" void kernel_launch(void* const* d_in, const int* in_sizes, int n_in,
                              void* d_out, int out_size, void* d_ws, size_t ws_size,
                              hipStream_t stream) {
  const float* s  = (const float*)d_in[0];  // s_out  [B, C] f32
  const float* t  = (const float*)d_in[1];  // t_out  [B, C] f32
  const int* pos  = (const int*)d_in[2];    // pos_idx [B, K] i32
  float* out_s    = (float*)d_out;          // output 0: s_out copy, B*C floats
  float* out_loss = out_s + (size_t)B_ * C_;// output 1: loss, B floats
  float* ws       = (float*)d_ws;           // 2*B floats of scratch

  kd_row_kernel<<<B_, TPB, 0, stream>>>(s, t, pos, out_s, ws);
  kd_final_kernel<<<1, TPB, 0, stream>>>(ws, out_loss);
}